// QuantizationCell_82617990906043
// MI455X (gfx1250) — compile-verified
//
#include <hip/hip_runtime.h>
#include <hip/hip_bf16.h>

// ---------------------------------------------------------------------------
// VQ-VAE quantization for MI455X (gfx1250, wave32, WMMA f32 16x16x4)
//   x:        (4, 128, 32, 32, 32) f32   -> flat (N=131072, D=128) where
//             flat[n,d] = x[b, d, s], n = b*32768 + s (spatial contiguous)
//   codebook: (512, 128) f32
// Outputs concatenated in d_out:
//   quantized_out (4,128,32,32,32) : 16,777,216 f32
//   encoding      (N, 512)         : 67,108,864 f32
//   distances     (N, 512)         : 67,108,864 f32
//   loss          scalar           : 1 f32  = 2 * sum(min_dist) / (N*D)
// ---------------------------------------------------------------------------

typedef __attribute__((ext_vector_type(2))) float v2f;
typedef __attribute__((ext_vector_type(8))) float v8f;

#define DIM   128
#define SPAT  32768          // 32*32*32
#define NCODE 512
#define NROWS 131072         // 4 * SPAT
#define ROWTILES 8192        // NROWS / 16
#define WAVES_PER_BLOCK 4
#define NBLOCKS (ROWTILES / WAVES_PER_BLOCK)   // 2048

// ---- codebook norms ‖w_k‖² -> ws[0..511] -----------------------------------
__global__ void vq_wnorm_kernel(const float* __restrict__ cb,
                                float* __restrict__ wnorm) {
    int t = blockIdx.x * blockDim.x + threadIdx.x;
    if (t < NCODE) {
        const float4* p = (const float4*)(cb + (size_t)t * DIM);
        float s = 0.f;
        #pragma unroll
        for (int i = 0; i < DIM / 4; ++i) {
            float4 v = p[i];
            s += v.x * v.x + v.y * v.y + v.z * v.z + v.w * v.w;
        }
        wnorm[t] = s;
    }
}

// ---- main kernel: distances + argmin + one-hot + gather + loss partials ----
__global__ __launch_bounds__(128) void vq_main_kernel(
    const float* __restrict__ x, const float* __restrict__ cb,
    const float* __restrict__ wnorm, float* __restrict__ qout,
    float* __restrict__ enc, float* __restrict__ dist,
    float* __restrict__ partials) {

    __shared__ float sval[WAVES_PER_BLOCK][16];
    __shared__ int   sidx[WAVES_PER_BLOCK][16];
    __shared__ float wsum[WAVES_PER_BLOCK];

    const int lane = threadIdx.x & 31;
    const int wave = threadIdx.x >> 5;
    const int lrow = lane & 15;          // row within 16-row tile (and col-in-tile)
    const int hi   = lane >> 4;          // half-wave selector

    const int tile = blockIdx.x * WAVES_PER_BLOCK + wave;   // 16-row tile id
    const int m0   = tile << 4;                             // first row
    const int b    = m0 >> 15;                              // batch (SPAT=2^15)
    const int s0   = m0 & (SPAT - 1);                       // spatial offset

    // ------------- load A tile (16 rows x 128 K) into 64 VGPRs -------------
    // ISA 16x4 f32 A layout: lane L(<16): M=L, VGPR0=K0, VGPR1=K1;
    //                        lane L+16 :  M=L, VGPR0=K2, VGPR1=K3.
    const float* xb = x + (size_t)b * DIM * SPAT + s0 + lrow;
    v2f a[32];
    float xn = 0.f;
    #pragma unroll
    for (int ko = 0; ko < 32; ++ko) {
        const int k = 4 * ko + 2 * hi;
        float a0 = xb[(size_t)k * SPAT];
        float a1 = xb[(size_t)(k + 1) * SPAT];
        a[ko].x = a0;
        a[ko].y = a1;
        xn += a0 * a0 + a1 * a1;
    }
    xn += __shfl_xor(xn, 16);            // full ‖x_row(lrow)‖² in every lane

    // xnr[r] = ‖x‖² of the row this lane's accumulator slot r refers to
    float xnr[8];
    #pragma unroll
    for (int r = 0; r < 8; ++r) xnr[r] = __shfl(xn, r + 8 * hi);

    float minv[8];
    int   mini[8];
    #pragma unroll
    for (int r = 0; r < 8; ++r) { minv[r] = 3.402823466e38f; mini[r] = 0; }

    // ------------- 32 column tiles of 16 codes each -------------
    #pragma unroll 1
    for (int ct = 0; ct < 32; ++ct) {
        const int n0 = ct << 4;
        // B (4x16) pair per K-step: lane L(<16): N=L, K0/K1; lane L+16: K2/K3
        const float* cbp = cb + (size_t)(n0 + lrow) * DIM + 2 * hi;
        v8f acc = {0.f, 0.f, 0.f, 0.f, 0.f, 0.f, 0.f, 0.f};
        #pragma unroll
        for (int ko = 0; ko < 32; ++ko) {
            const float2 t = *(const float2*)(cbp + 4 * ko);
            v2f bv;
            bv.x = t.x;
            bv.y = t.y;
            acc = __builtin_amdgcn_wmma_f32_16x16x4_f32(
                false, a[ko], false, bv, (short)0, acc, false, false);
        }
        const float wn = wnorm[n0 + lrow];
        const int   n  = n0 + lrow;
        #pragma unroll
        for (int r = 0; r < 8; ++r) {
            // acc slot r = element (M = r + 8*hi, N = lrow)
            float dv = xnr[r] - 2.0f * acc[r] + wn;
            dist[(size_t)(m0 + r + 8 * hi) * NCODE + n] = dv;
            if (dv < minv[r]) { minv[r] = dv; mini[r] = n; }  // n increases -> first idx kept
        }
    }

    // ------------- argmin across the 16 lanes of each half-wave -------------
    #pragma unroll
    for (int r = 0; r < 8; ++r) {
        #pragma unroll
        for (int st = 1; st < 16; st <<= 1) {
            float ov = __shfl_xor(minv[r], st);
            int   oi = __shfl_xor(mini[r], st);
            if (ov < minv[r] || (ov == minv[r] && oi < mini[r])) {
                minv[r] = ov;
                mini[r] = oi;
            }
        }
    }
    if (lane == 0) {
        #pragma unroll
        for (int r = 0; r < 8; ++r) { sval[wave][r] = minv[r]; sidx[wave][r] = mini[r]; }
    }
    if (lane == 16) {
        #pragma unroll
        for (int r = 0; r < 8; ++r) { sval[wave][8 + r] = minv[r]; sidx[wave][8 + r] = mini[r]; }
    }
    __syncthreads();

    // ------------- loss partial (sum of 16 min distances per wave) ----------
    float lv = (lane < 16) ? sval[wave][lane] : 0.f;
    #pragma unroll
    for (int st = 1; st < 32; st <<= 1) lv += __shfl_xor(lv, st);
    if (lane == 0) wsum[wave] = lv;

    // ------------- one-hot encoding rows (coalesced 512B bursts) ------------
    float* erow = enc + (size_t)m0 * NCODE;
    #pragma unroll 1
    for (int r = 0; r < 16; ++r) {
        const int id = sidx[wave][r];
        float4* ep = (float4*)(erow + (size_t)r * NCODE);
        #pragma unroll
        for (int j = 0; j < 4; ++j) {
            const int c = j * 128 + lane * 4;
            float4 v;
            v.x = (c + 0 == id) ? 1.f : 0.f;
            v.y = (c + 1 == id) ? 1.f : 0.f;
            v.z = (c + 2 == id) ? 1.f : 0.f;
            v.w = (c + 3 == id) ? 1.f : 0.f;
            ep[j * 32 + lane] = v;
        }
    }

    // ------------- quantized output, channel-first layout -------------------
    // lane = lrow + 16*hi handles row lrow, channels [64*hi, 64*hi+64)
    {
        const int id = sidx[wave][lrow];
        const float* crow = cb + (size_t)id * DIM;
        float* qb = qout + (size_t)b * DIM * SPAT + s0 + lrow;
        #pragma unroll 1
        for (int j = 0; j < 64; ++j) {
            const int d = 64 * hi + j;
            qb[(size_t)d * SPAT] = crow[d];
        }
    }

    __syncthreads();
    if (threadIdx.x == 0)
        partials[blockIdx.x] = wsum[0] + wsum[1] + wsum[2] + wsum[3];
}

// ---- deterministic fixed-order loss reduction ------------------------------
__global__ void vq_loss_kernel(const float* __restrict__ partials,
                               float* __restrict__ lossout) {
    __shared__ float sm[256];
    float s = 0.f;
    for (int i = threadIdx.x; i < NBLOCKS; i += 256) s += partials[i];
    sm[threadIdx.x] = s;
    __syncthreads();
    for (int st = 128; st > 0; st >>= 1) {
        if (threadIdx.x < st) sm[threadIdx.x] += sm[threadIdx.x + st];
        __syncthreads();
    }
    if (threadIdx.x == 0)
        lossout[0] = 2.0f * sm[0] / 16777216.0f;   // 2 * sum / (N*D)
}

extern "C" void kernel_launch(void* const* d_in, const int* in_sizes, int n_in,
                              void* d_out, int out_size, void* d_ws, size_t ws_size,
                              hipStream_t stream) {
    const float* x  = (const float*)d_in[0];
    const float* cb = (const float*)d_in[1];

    float* qout  = (float*)d_out;                    // 16,777,216
    float* enc   = qout + (size_t)16777216;          // 67,108,864
    float* dist  = enc + (size_t)67108864;           // 67,108,864
    float* lossp = dist + (size_t)67108864;          // 1

    float* wnorm    = (float*)d_ws;                  // 512 floats
    float* partials = wnorm + NCODE;                 // NBLOCKS floats

    vq_wnorm_kernel<<<2, 256, 0, stream>>>(cb, wnorm);
    vq_main_kernel<<<NBLOCKS, 128, 0, stream>>>(x, cb, wnorm, qout, enc, dist,
                                                partials);
    vq_loss_kernel<<<1, 256, 0, stream>>>(partials, lossp);
}